// WindowSlice_81389630259937
// MI455X (gfx1250) — compile-verified
//
#include <hip/hip_runtime.h>
#include <hip/hip_bf16.h>

// Problem constants from the reference:
//   B = 64, T = 4096, C = 128, L = ceil(0.9*T) = 3687
#define B_DIM    64
#define T_DIM    4096
#define C_DIM    128
#define L_DIM    3687
#define T_TILE   64          // output rows per workgroup
#define N_TILES  (T_DIM / T_TILE)          // 64
#define ROWS_MAX 64          // LDS staging rows (max true window = 59)
#define BLOCK    256         // 8 wave32

typedef float v4f __attribute__((ext_vector_type(4)));   // native clang vector

// q[t] = t * L/(T-1); clamp; split into integer row + fractional weight.
__device__ __forceinline__ int lo_of(int t, float& w) {
    const float STEP = (float)((double)L_DIM / (double)(T_DIM - 1)); // 3687/4095
    float q  = (float)t * STEP;
    float tt = fminf(q, (float)(L_DIM - 1));       // q >= 0 always
    int lo   = (int)tt;                            // trunc == floor (non-negative)
    if (lo > L_DIM - 2) lo = L_DIM - 2;
    w = tt - (float)lo;
    return lo;
}

__global__ __launch_bounds__(BLOCK)
void WindowSlice_81389630259937_kernel(const float* __restrict__ x,
                                       const int*   __restrict__ starts,
                                       float*       __restrict__ out) {
    __shared__ __align__(16) float smem[ROWS_MAX * C_DIM];   // 32 KB

    const int tid  = threadIdx.x;
    const int wg   = blockIdx.x;
    const int b    = wg / N_TILES;
    const int tile = wg - b * N_TILES;
    const int t0   = tile * T_TILE;

    const int start = starts[b];              // uniform scalar load

    // Input-row window for this tile (lo monotone in t => contiguous coverage).
    float wdum;
    const int lo_first = lo_of(t0, wdum);
    const int lo_last  = lo_of(t0 + T_TILE - 1, wdum);
    int nrows = lo_last + 2 - lo_first;       // includes the +1 "hi" row
    if (nrows > ROWS_MAX) nrows = ROWS_MAX;   // provably <= 59; safety only

    // Contiguous global region to stage: nrows * 512 bytes.
    const float* gbase = x + ((size_t)b * T_DIM + (size_t)(start + lo_first)) * C_DIM;
    const unsigned lds0 = (unsigned)(size_t)(void*)smem;  // flat->LDS: low 32 bits
    const int nbytes = nrows * (C_DIM * 4);

    // CDNA5 async DMA: global -> LDS, 16 B per lane-issue, GVS addressing
    // (SGPR 64-bit base + per-lane 32-bit byte offset), ASYNCcnt-tracked.
    // Input rows have no cross-workgroup reuse -> non-temporal load hint.
    for (int ofs = tid * 16; ofs < nbytes; ofs += BLOCK * 16) {
        unsigned lds_addr = lds0 + (unsigned)ofs;
        unsigned goff     = (unsigned)ofs;
        asm volatile("global_load_async_to_lds_b128 %0, %1, %2 th:TH_LOAD_NT"
                     :
                     : "v"(lds_addr), "v"(goff), "s"(gbase)
                     : "memory");
    }
    // Each wave drains its own async queue; the workgroup barrier then makes
    // every wave's staged rows visible to all waves.
    asm volatile("s_wait_asynccnt 0x0" ::: "memory");
    __syncthreads();

    // 8 rows per pass (one row per wave32), T_TILE/8 passes.
    const int lane = tid & 31;
    const int wid  = tid >> 5;
    #pragma unroll
    for (int pass = 0; pass < T_TILE / 8; ++pass) {
        const int r = pass * 8 + wid;
        const int t = t0 + r;
        float w;
        const int lo = lo_of(t, w);
        const int lr = lo - lo_first;         // row inside the LDS window

        const v4f vlo = *reinterpret_cast<const v4f*>(&smem[lr * C_DIM + lane * 4]);
        const v4f vhi = *reinterpret_cast<const v4f*>(&smem[(lr + 1) * C_DIM + lane * 4]);

        const float om = 1.0f - w;            // match v_lo*(1-w) + v_hi*w exactly
        v4f o;
        o.x = vlo.x * om + vhi.x * w;
        o.y = vlo.y * om + vhi.y * w;
        o.z = vlo.z * om + vhi.z * w;
        o.w = vlo.w * om + vhi.w * w;

        // Output is write-once; non-temporal store keeps L2 for the input stream.
        __builtin_nontemporal_store(
            o, reinterpret_cast<v4f*>(&out[((size_t)b * T_DIM + t) * C_DIM + lane * 4]));
    }
}

extern "C" void kernel_launch(void* const* d_in, const int* in_sizes, int n_in,
                              void* d_out, int out_size, void* d_ws, size_t ws_size,
                              hipStream_t stream) {
    const float* x      = (const float*)d_in[0];
    const int*   starts = (const int*)d_in[1];
    float*       out    = (float*)d_out;

    const int n_wgs = B_DIM * N_TILES;   // 64 * 64 = 4096
    WindowSlice_81389630259937_kernel<<<n_wgs, BLOCK, 0, stream>>>(x, starts, out);
}